// IPFECNN_47880295415814
// MI455X (gfx1250) — compile-verified
//
#include <hip/hip_runtime.h>
#include <hip/hip_bf16.h>

#define P_PRIME 1000003u
#define NORD    1000002u   // p - 1
#define IP_MAX  9000       // |<x,y>| <= 9*20*50

typedef __attribute__((ext_vector_type(2))) float v2f;
typedef __attribute__((ext_vector_type(8))) float v8f;

// ---------------- modular arithmetic ----------------
// all exponents < 2^20, fixed trip count (no divergent while)
__device__ __forceinline__ unsigned int modpow_u(unsigned int b, unsigned int e) {
    unsigned long long r = 1ull, bb = (unsigned long long)(b % P_PRIME);
    for (int i = 0; i < 20; ++i) {
        if (e & 1u) r = (r * bb) % P_PRIME;
        bb = (bb * bb) % P_PRIME;
        e >>= 1;
    }
    return (unsigned int)r;
}

// ---------------- misc ----------------
__global__ void k_zero_f32(float* p, int n) {
    int i = blockIdx.x * blockDim.x + threadIdx.x;
    if (i < n) p[i] = 0.f;
}

// ---------------- dlog table ----------------
__global__ void k_tab_init(int* tab) {
    int i = blockIdx.x * blockDim.x + threadIdx.x;
    if (i < (int)P_PRIME) tab[i] = 0;
}

__global__ void k_tab_build(int* tab, const int* gp) {
    int t = blockIdx.x * blockDim.x + threadIdx.x;
    if (t >= 2 * IP_MAX + 1) return;
    unsigned int g = (unsigned int)gp[0];
    int e = t - IP_MAX;
    unsigned int eu = (e < 0) ? (unsigned int)(e + (int)NORD) : (unsigned int)e;
    tab[modpow_u(g, eu)] = e;
}

// ---------------- IPFE decrypt -> conv1 feature maps ----------------
__global__ void k_decrypt(const int* tab, const int* ct0, const int* cts,
                          const int* y, const int* sk_y, const float* bias1,
                          float* feat1) {
    int tid = blockIdx.x * blockDim.x + threadIdx.x;
    if (tid >= 16 * 4 * 784) return;
    int k = tid / (4 * 784);
    int r = tid % (4 * 784);
    int b = r / 784;
    int j = r % 784;
    unsigned long long num = 1ull;
    for (int i = 0; i < 9; ++i) {
        int yv = y[k * 9 + i] % (int)NORD;
        if (yv < 0) yv += (int)NORD;
        unsigned int c = (unsigned int)cts[(b * 784 + j) * 9 + i];
        num = (num * modpow_u(c, (unsigned int)yv)) % P_PRIME;
    }
    unsigned int den = modpow_u((unsigned int)ct0[b * 784 + j], (unsigned int)sk_y[k]);
    unsigned int inv = modpow_u(den, P_PRIME - 2u);        // Fermat inverse
    unsigned int val = (unsigned int)((num * inv) % P_PRIME);
    int ip = tab[val];
    feat1[(b * 16 + k) * 784 + j] = (float)ip * 1e-4f + bias1[k];
}

// ---------------- stage 1: BN + ReLU + 2x2 maxpool (28->14) ----------------
__global__ void k_bn_relu_pool1(const float* feat1, const float* g_, const float* b_,
                                const float* m_, const float* v_, float* x1) {
    int tid = blockIdx.x * blockDim.x + threadIdx.x;
    if (tid >= 4 * 16 * 14 * 14) return;
    int ow = tid % 14; int t = tid / 14;
    int oh = t % 14;   t /= 14;
    int c  = t % 16;   int b = t / 16;
    float s = g_[c] * rsqrtf(v_[c] + 1e-5f);
    float mm = m_[c], bb = b_[c];
    const float* base = feat1 + (b * 16 + c) * 784;
    float best = 0.f;   // ReLU outputs are >= 0
    for (int dy = 0; dy < 2; ++dy)
        for (int dx = 0; dx < 2; ++dx) {
            float v0 = (base[(2 * oh + dy) * 28 + (2 * ow + dx)] - mm) * s + bb;
            best = fmaxf(best, fmaxf(v0, 0.f));
        }
    x1[(b * 16 + c) * 196 + oh * 14 + ow] = best;
}

// ---------------- im2col (padded rows are zero-filled) ----------------
// conv2: x1[4,16,14,14] -> A2[784,144]   (784 = 49*16, no pad needed)
__global__ void k_im2col2(const float* x1, float* A2) {
    int tid = blockIdx.x * blockDim.x + threadIdx.x;
    if (tid >= 784 * 144) return;
    int m = tid / 144, k = tid % 144;
    int b = m / 196, hw = m % 196;
    int h = hw / 14, w = hw % 14;
    int c = k / 9, rr = k % 9;
    int kh = rr / 3, kw = rr % 3;
    int ih = h + kh - 1, iw = w + kw - 1;
    float v = 0.f;
    if (ih >= 0 && ih < 14 && iw >= 0 && iw < 14)
        v = x1[(b * 16 + c) * 196 + ih * 14 + iw];
    A2[m * 144 + k] = v;
}

// conv3: x2[4,32,7,7] -> A3[208,288]  (rows 196..207 zero)
__global__ void k_im2col3(const float* x2, float* A3) {
    int tid = blockIdx.x * blockDim.x + threadIdx.x;
    if (tid >= 208 * 288) return;
    int m = tid / 288, k = tid % 288;
    float v = 0.f;
    if (m < 196) {
        int b = m / 49, hw = m % 49;
        int h = hw / 7, w = hw % 7;
        int c = k / 9, rr = k % 9;
        int kh = rr / 3, kw = rr % 3;
        int ih = h + kh - 1, iw = w + kw - 1;
        if (ih >= 0 && ih < 7 && iw >= 0 && iw < 7)
            v = x2[(b * 32 + c) * 49 + ih * 7 + iw];
    }
    A3[m * 288 + k] = v;
}

// ---------------- B pack: pair-interleave K, pad N to Np ----------------
// Bp[(k/2)*(Np*2) + n*2 + (k&1)] = B[k][n]  (0 beyond N)
__global__ void k_packB(const float* src, int bsk, int bsn, int K, int N, int Np,
                        float* Bp) {
    int tid = blockIdx.x * blockDim.x + threadIdx.x;
    if (tid >= K * Np) return;
    int k = tid / Np, n = tid % Np;
    float v = (n < N) ? src[k * bsk + n * bsn] : 0.f;
    Bp[(k >> 1) * (Np * 2) + n * 2 + (k & 1)] = v;
}

// ---------------- f32 WMMA GEMM, unconditional inner loop ----------------
// A: padded to tilesM*16 rows (zeros beyond Mstore). Bp: packed as above.
__global__ void k_gemm_wmma(const float* __restrict__ A, int lda,
                            const float* __restrict__ Bp, int Np,
                            int Mstore, int Nstore, int K,
                            const float* bias, const float* bng, const float* bnb,
                            const float* bnm, const float* bnv, int relu,
                            float* __restrict__ C, int ldc,
                            int tilesN, int totalTiles) {
    int wave = (blockIdx.x * blockDim.x + threadIdx.x) >> 5;
    if (wave >= totalTiles) return;            // wave-uniform exit
    int lane = threadIdx.x & 31;
    int tm = wave / tilesN, tn = wave % tilesN;
    int half = lane >> 4;                      // 0: lanes 0-15, 1: lanes 16-31
    int l16  = lane & 15;
    int row0 = tm * 16, col0 = tn * 16;
    // per-lane streaming pointers (8-byte aligned: lda, Np even)
    const float* ap = A + (row0 + l16) * lda + 2 * half;        // A[row][ka..ka+1]
    const float* bp = Bp + half * (Np * 2) + (col0 + l16) * 2;  // {B[ka][col],B[ka+1][col]}
    v8f acc = {};
#pragma unroll 4
    for (int k0 = 0; k0 < K; k0 += 4) {
        v2f a = *(const v2f*)ap;
        v2f b = *(const v2f*)bp;
        ap += 4;
        bp += 4 * Np;
        acc = __builtin_amdgcn_wmma_f32_16x16x4_f32(
            /*neg_a=*/false, a, /*neg_b=*/false, b,
            /*c_mod=*/(short)0, acc, /*reuse_a=*/false, /*reuse_b=*/false);
    }
    // fused epilogue (per-output-channel = per-column)
    int col = col0 + l16;
    float bb = 0.f, sc = 1.f, mu = 0.f, sh = 0.f;
    if (col < Nstore) {
        if (bias) bb = bias[col];
        if (bng) { sc = bng[col] * rsqrtf(bnv[col] + 1e-5f); mu = bnm[col]; sh = bnb[col]; }
    }
    for (int v = 0; v < 8; ++v) {
        int row = row0 + v + 8 * half;         // D layout: VGPR v -> M=v (+8 for hi half)
        if (row < Mstore && col < Nstore) {
            float o = acc[v] + bb;
            if (bng) o = (o - mu) * sc + sh;
            if (relu) o = fmaxf(o, 0.f);
            C[row * ldc + col] = o;
        }
    }
}

// ---------------- pool after conv2: G2[784,32] -> x2[4,32,7,7] ----------------
__global__ void k_pool2(const float* G2, float* x2) {
    int tid = blockIdx.x * blockDim.x + threadIdx.x;
    if (tid >= 4 * 32 * 7 * 7) return;
    int ow = tid % 7; int t = tid / 7;
    int oh = t % 7;   t /= 7;
    int c  = t % 32;  int b = t / 32;
    float best = 0.f;                          // inputs are post-ReLU
    for (int dy = 0; dy < 2; ++dy)
        for (int dx = 0; dx < 2; ++dx)
            best = fmaxf(best, G2[(b * 196 + (2 * oh + dy) * 14 + (2 * ow + dx)) * 32 + c]);
    x2[(b * 32 + c) * 49 + oh * 7 + ow] = best;
}

// ---------------- pool after conv3: G3[196,64] -> x3 flat [4,576] -------------
__global__ void k_pool3(const float* G3, float* x3) {
    int tid = blockIdx.x * blockDim.x + threadIdx.x;
    if (tid >= 4 * 64 * 3 * 3) return;
    int ow = tid % 3; int t = tid / 3;
    int oh = t % 3;   t /= 3;
    int c  = t % 64;  int b = t / 64;
    float best = 0.f;
    for (int dy = 0; dy < 2; ++dy)
        for (int dx = 0; dx < 2; ++dx)
            best = fmaxf(best, G3[(b * 49 + (2 * oh + dy) * 7 + (2 * ow + dx)) * 64 + c]);
    x3[b * 576 + c * 9 + oh * 3 + ow] = best;
}

// ---------------- workspace layout (bytes, 256-aligned) ----------------
#define OFF_TAB   0u          // 1,000,003 int32
#define OFF_FEAT1 4000256u    // 4*16*28*28 f32 = 200,704 B
#define OFF_X1    4200960u    // 4*16*14*14 f32 = 50,176 B
#define OFF_A2    4251136u    // 784*144 f32 = 451,584 B
#define OFF_BP2   4702720u    // 144*32 f32 = 18,432 B
#define OFF_G2    4721152u    // 784*32 f32 = 100,352 B
#define OFF_X2    4821504u    // 4*32*7*7 f32 = 25,088 B
#define OFF_A3    4846592u    // 208*288 f32 = 239,616 B
#define OFF_BP3   5086208u    // 288*64 f32 = 73,728 B
#define OFF_G3    5159936u    // 196*64 f32 = 50,176 B
#define OFF_X3    5210112u    // 16*576 f32 = 36,864 B (rows 4..15 zero)
#define OFF_BPF1  5246976u    // 576*128 f32 = 294,912 B
#define OFF_H1    5541888u    // 16*128 f32 = 8,192 B (rows 4..15 zero)
#define OFF_BPF2  5550080u    // 128*16 f32 = 8,192 B

extern "C" void kernel_launch(void* const* d_in, const int* in_sizes, int n_in,
                              void* d_out, int out_size, void* d_ws, size_t ws_size,
                              hipStream_t stream) {
    const int*   ct0    = (const int*)d_in[0];
    const int*   cts    = (const int*)d_in[1];
    const int*   y      = (const int*)d_in[2];
    const int*   sk_y   = (const int*)d_in[3];
    const float* bias1  = (const float*)d_in[4];
    const float* bn1g   = (const float*)d_in[5];
    const float* bn1b   = (const float*)d_in[6];
    const float* bn1m   = (const float*)d_in[7];
    const float* bn1v   = (const float*)d_in[8];
    const float* conv2w = (const float*)d_in[9];
    const float* conv2b = (const float*)d_in[10];
    const float* bn2g   = (const float*)d_in[11];
    const float* bn2b   = (const float*)d_in[12];
    const float* bn2m   = (const float*)d_in[13];
    const float* bn2v   = (const float*)d_in[14];
    const float* conv3w = (const float*)d_in[15];
    const float* conv3b = (const float*)d_in[16];
    const float* bn3g   = (const float*)d_in[17];
    const float* bn3b   = (const float*)d_in[18];
    const float* bn3m   = (const float*)d_in[19];
    const float* bn3v   = (const float*)d_in[20];
    const float* fc1w   = (const float*)d_in[21];
    const float* fc1b   = (const float*)d_in[22];
    const float* fc2w   = (const float*)d_in[23];
    const float* fc2b   = (const float*)d_in[24];
    const int*   gp     = (const int*)d_in[25];   // generator g ; p is hardcoded

    char*  ws    = (char*)d_ws;
    int*   tab   = (int*)  (ws + OFF_TAB);
    float* feat1 = (float*)(ws + OFF_FEAT1);
    float* x1    = (float*)(ws + OFF_X1);
    float* A2    = (float*)(ws + OFF_A2);
    float* BP2   = (float*)(ws + OFF_BP2);
    float* G2    = (float*)(ws + OFF_G2);
    float* x2    = (float*)(ws + OFF_X2);
    float* A3    = (float*)(ws + OFF_A3);
    float* BP3   = (float*)(ws + OFF_BP3);
    float* G3    = (float*)(ws + OFF_G3);
    float* x3    = (float*)(ws + OFF_X3);
    float* BPF1  = (float*)(ws + OFF_BPF1);
    float* h1    = (float*)(ws + OFF_H1);
    float* BPF2  = (float*)(ws + OFF_BPF2);
    float* out   = (float*)d_out;

    // 1) dlog lookup table
    k_tab_init <<<(P_PRIME + 255) / 256, 256, 0, stream>>>(tab);
    k_tab_build<<<(2 * IP_MAX + 1 + 255) / 256, 256, 0, stream>>>(tab, gp);

    // weight packs (independent of data path)
    k_packB<<<(144 * 32 + 255) / 256, 256, 0, stream>>>(conv2w, 1, 144, 144, 32, 32, BP2);
    k_packB<<<(288 * 64 + 255) / 256, 256, 0, stream>>>(conv3w, 1, 288, 288, 64, 64, BP3);
    k_packB<<<(576 * 128 + 255) / 256, 256, 0, stream>>>(fc1w, 1, 576, 576, 128, 128, BPF1);
    k_packB<<<(128 * 16 + 255) / 256, 256, 0, stream>>>(fc2w, 1, 128, 128, 10, 16, BPF2);
    // zero padded A buffers for FC stages
    k_zero_f32<<<(16 * 576 + 255) / 256, 256, 0, stream>>>(x3, 16 * 576);
    k_zero_f32<<<(16 * 128 + 255) / 256, 256, 0, stream>>>(h1, 16 * 128);

    // 2) IPFE decrypt -> feat1 [4,16,28,28]
    k_decrypt<<<(16 * 4 * 784 + 255) / 256, 256, 0, stream>>>(
        tab, ct0, cts, y, sk_y, bias1, feat1);

    // 3) BN1 + ReLU + pool -> x1 [4,16,14,14]
    k_bn_relu_pool1<<<(4 * 16 * 14 * 14 + 255) / 256, 256, 0, stream>>>(
        feat1, bn1g, bn1b, bn1m, bn1v, x1);

    // 4) conv2 as WMMA GEMM: [784,144]x[144,32], fused bias+BN2+ReLU
    k_im2col2<<<(784 * 144 + 255) / 256, 256, 0, stream>>>(x1, A2);
    {
        int tiles = 49 * 2;
        k_gemm_wmma<<<(tiles + 3) / 4, 128, 0, stream>>>(
            A2, 144, BP2, 32, 784, 32, 144,
            conv2b, bn2g, bn2b, bn2m, bn2v, /*relu=*/1, G2, 32, 2, tiles);
    }
    k_pool2<<<(4 * 32 * 7 * 7 + 255) / 256, 256, 0, stream>>>(G2, x2);

    // 5) conv3 as WMMA GEMM: [208,288]x[288,64], fused bias+BN3+ReLU
    k_im2col3<<<(208 * 288 + 255) / 256, 256, 0, stream>>>(x2, A3);
    {
        int tiles = 13 * 4;
        k_gemm_wmma<<<(tiles + 3) / 4, 128, 0, stream>>>(
            A3, 288, BP3, 64, 196, 64, 288,
            conv3b, bn3g, bn3b, bn3m, bn3v, 1, G3, 64, 4, tiles);
    }
    k_pool3<<<(4 * 64 * 3 * 3 + 255) / 256, 256, 0, stream>>>(G3, x3);

    // 6) fc1: [16(pad4),576]x[576,128] (+bias, ReLU) -> h1[16,128]
    {
        int tiles = 1 * 8;
        k_gemm_wmma<<<(tiles + 3) / 4, 128, 0, stream>>>(
            x3, 576, BPF1, 128, 4, 128, 576,
            fc1b, nullptr, nullptr, nullptr, nullptr, 1, h1, 128, 8, tiles);
    }
    // 7) fc2: [16(pad4),128]x[128,10(pad16)] (+bias) -> d_out [4,10]
    {
        int tiles = 1;
        k_gemm_wmma<<<1, 128, 0, stream>>>(
            h1, 128, BPF2, 16, 4, 10, 128,
            fc2b, nullptr, nullptr, nullptr, nullptr, 0, out, 10, 1, tiles);
    }
}